// CRF_52905407152797
// MI455X (gfx1250) — compile-verified
//
#include <hip/hip_runtime.h>
#include <stdint.h>

typedef __attribute__((ext_vector_type(2))) float v2f;
typedef __attribute__((ext_vector_type(8))) float v8f;

#define BB 128
#define SS 512
#define UU 1024
#define TT 48
#define MROWS (BB * SS)
#define NCHUNK (UU / 64)

// Async global -> LDS 16-byte transfer (GLOBAL_LOAD_ASYNC_TO_LDS_B128, GV mode).
// lds_off: wave-relative LDS byte address (low 32 bits of flat shared pointer).
// gptr:    64-bit global address (VGPR pair).
__device__ __forceinline__ void async_ld16(uint32_t lds_off, const void* gptr) {
    asm volatile("global_load_async_to_lds_b128 %0, %1, off"
                 :: "v"(lds_off), "v"(gptr) : "memory");
}

// ---------------------------------------------------------------------------
// Kernel 1: scores[M=65536, T=48] = H[M, U=1024] @ W[U, 48] + b
// FP32 WMMA 16x16x4, double-buffered async global->LDS staging (ASYNCcnt).
// 128 threads: waves 0-2 compute one 16-wide N tile each; all 4 waves load.
// ---------------------------------------------------------------------------
__global__ __launch_bounds__(128) void crf_scores_wmma(
    const float* __restrict__ H, const float* __restrict__ W,
    const float* __restrict__ bias, float* __restrict__ scores)
{
    __shared__ __attribute__((aligned(16))) float Hs[2][16][64];  // 8 KB
    __shared__ __attribute__((aligned(16))) float Ws[2][64][TT];  // 24 KB

    const int tid  = threadIdx.x;
    const int lane = tid & 31;
    const int wave = tid >> 5;         // 0..3; waves 0-2 compute
    const int nb   = wave * 16;
    const int col  = lane & 15;        // n within tile (B/D); also m row for A
    const int kh   = lane >> 4;        // lane-half selects K pair
    const long rowBase = (long)blockIdx.x * 16;

    // Issue exactly 8 async B128 transfers per thread for K-chunk k0 -> buf.
    auto issue_chunk = [&](int k0, int buf) {
        #pragma unroll
        for (int j = 0; j < 2; ++j) {            // H tile: 256 float4
            const int idx = tid + j * 128;
            const int m = idx >> 4, k4 = idx & 15;
            async_ld16((uint32_t)(uintptr_t)&Hs[buf][m][k4 * 4],
                       H + (rowBase + m) * UU + k0 + k4 * 4);
        }
        const float4* src = (const float4*)(W + (long)k0 * TT);
        float4*       dst = (float4*)&Ws[buf][0][0];
        #pragma unroll
        for (int j = 0; j < 6; ++j) {            // W chunk: 768 float4 contiguous
            const int idx = tid + j * 128;
            async_ld16((uint32_t)(uintptr_t)(dst + idx), src + idx);
        }
    };

    v8f acc = {};
    issue_chunk(0, 0);

    for (int c = 0; c < NCHUNK; ++c) {
        const int cur = c & 1;
        if (c + 1 < NCHUNK) {
            issue_chunk((c + 1) * 64, cur ^ 1);
            asm volatile("s_wait_asynccnt 0x8" ::: "memory"); // prev 8 landed
        } else {
            asm volatile("s_wait_asynccnt 0x0" ::: "memory");
        }
        __syncthreads();   // all waves' async data for buf[cur] visible

        if (wave < 3) {    // wave-uniform: EXEC all-ones inside
            #pragma unroll
            for (int kk = 0; kk < 64; kk += 4) {
                v2f a, bf;
                a.x  = Hs[cur][col][kk + 2 * kh + 0];        // A: M=col, K=2*kh+j
                a.y  = Hs[cur][col][kk + 2 * kh + 1];
                bf.x = Ws[cur][kk + 2 * kh + 0][nb + col];   // B: K=2*kh+j, N=col
                bf.y = Ws[cur][kk + 2 * kh + 1][nb + col];
                acc = __builtin_amdgcn_wmma_f32_16x16x4_f32(
                    false, a, false, bf, (short)0, acc, false, false);
            }
        }
        __syncthreads();   // reads of buf[cur] done before it is refilled
    }

    if (wave < 3) {
        const float bval = bias[nb + col];
        const int   mset = (lane < 16) ? 0 : 8;
        #pragma unroll
        for (int r = 0; r < 8; ++r) {
            scores[(rowBase + mset + r) * TT + nb + col] = acc[r] + bval;
        }
    }
}

// ---------------------------------------------------------------------------
// Kernel 2: numerator per batch (gold-path score)
// ---------------------------------------------------------------------------
__global__ __launch_bounds__(64) void crf_numerator(
    const float* __restrict__ scores, const float* __restrict__ start_t,
    const float* __restrict__ end_t, const float* __restrict__ trans,
    const int* __restrict__ tag, const int* __restrict__ s_len,
    const float* __restrict__ w_mask, float* __restrict__ num)
{
    __shared__ float red[64];
    const int b   = blockIdx.x;
    const int tid = threadIdx.x;
    const int base = b * SS;

    float acc = 0.f;
    for (int s = tid; s < SS; s += 64) {
        const float msk = w_mask[base + s];
        const int   tg  = tag[base + s];
        acc += scores[(long)(base + s) * TT + tg] * msk;
        if (s > 0) {
            const int tp = tag[base + s - 1];
            acc += trans[tp * TT + tg] * msk;
        }
    }
    red[tid] = acc;
    __syncthreads();
    for (int off = 32; off > 0; off >>= 1) {
        if (tid < off) red[tid] += red[tid + off];
        __syncthreads();
    }
    if (tid == 0) {
        const int t0   = tag[base];
        const int last = tag[base + s_len[b] - 1];
        num[b] = red[0] + start_t[t0] + end_t[last];
    }
}

// ---------------------------------------------------------------------------
// Kernel 3: forward log-sum-exp scan + final log_Z; one block per batch,
// 48 active lanes (one per target state). Transitions transposed in LDS.
// ---------------------------------------------------------------------------
__global__ __launch_bounds__(64) void crf_scan(
    const float* __restrict__ scores, const float* __restrict__ start_t,
    const float* __restrict__ end_t, const float* __restrict__ trans,
    const float* __restrict__ w_mask, const float* __restrict__ num,
    float* __restrict__ out)
{
    __shared__ float Tt[TT][49];   // Tt[t'][t] = trans[t][t'], padded
    __shared__ float lp[TT];
    __shared__ float red[TT];

    const int b   = blockIdx.x;
    const int tid = threadIdx.x;
    const long sb = (long)b * SS;

    for (int idx = tid; idx < TT * TT; idx += 64) {
        const int t = idx / TT, tp = idx % TT;
        Tt[tp][t] = trans[t * TT + tp];
    }
    if (tid < TT) lp[tid] = start_t[tid] + scores[sb * TT + tid];
    __syncthreads();

    for (int i = 1; i < SS; ++i) {
        const float msk = w_mask[sb + i];
        float nv = 0.f;
        if (tid < TT) {
            const float e = scores[(sb + i) * TT + tid];
            float m = -3.4e38f;
            #pragma unroll 8
            for (int t = 0; t < TT; ++t)
                m = fmaxf(m, lp[t] + Tt[tid][t]);
            float ssum = 0.f;
            #pragma unroll 8
            for (int t = 0; t < TT; ++t)
                ssum += __expf(lp[t] + Tt[tid][t] - m);
            nv = m + __logf(ssum) + e;
        }
        __syncthreads();
        if (tid < TT && msk > 0.f) lp[tid] = nv;
        __syncthreads();
    }

    if (tid < TT) red[tid] = lp[tid] + end_t[tid];
    __syncthreads();
    if (tid == 0) {
        float m = -3.4e38f;
        for (int t = 0; t < TT; ++t) m = fmaxf(m, red[t]);
        float ssum = 0.f;
        for (int t = 0; t < TT; ++t) ssum += __expf(red[t] - m);
        out[b] = num[b] - (m + __logf(ssum));
    }
}

// ---------------------------------------------------------------------------
extern "C" void kernel_launch(void* const* d_in, const int* in_sizes, int n_in,
                              void* d_out, int out_size, void* d_ws, size_t ws_size,
                              hipStream_t stream) {
    (void)in_sizes; (void)n_in; (void)out_size; (void)ws_size;
    const float* H       = (const float*)d_in[0];
    const float* W       = (const float*)d_in[1];
    const float* bias    = (const float*)d_in[2];
    const float* start_t = (const float*)d_in[3];
    const float* end_t   = (const float*)d_in[4];
    const float* trans   = (const float*)d_in[5];
    const int*   tag     = (const int*)d_in[6];
    const int*   s_len   = (const int*)d_in[7];
    const float* w_mask  = (const float*)d_in[8];
    float*       out     = (float*)d_out;

    float* scores = (float*)d_ws;                       // 65536*48 floats
    float* num    = scores + (size_t)MROWS * TT;        // 128 floats

    crf_scores_wmma<<<MROWS / 16, 128, 0, stream>>>(H, W, bias, scores);
    crf_numerator<<<BB, 64, 0, stream>>>(scores, start_t, end_t, trans,
                                         tag, s_len, w_mask, num);
    crf_scan<<<BB, 64, 0, stream>>>(scores, start_t, end_t, trans,
                                    w_mask, num, out);
}